// iRPE_65180423685334
// MI455X (gfx1250) — compile-verified
//
#include <hip/hip_runtime.h>
#include <hip/hip_bf16.h>

// iRPE (PRODUCT, contextual/transposed):
//   Phase 1: lt[bh,l,k] = sum_d x[bh,l,d] * W[h,d,k]   (64 GEMMs 1024x64x49, fp32)
//            V_WMMA_F32_16X16X4_F32, W staged transposed+zero-padded in LDS.
//   Phase 2: out[bh,i,j] = lt[bh,i, rp[i,j]]           (256 MB store-bound gather)
// Roofline: phase 2 writes 256 MB @ 23.3 TB/s ~= 11 us; phase 1 is ~0.4 GFLOP (noise).

#define LQ   1024   // sequence length (32*32)
#define DD   64     // head dim
#define NB   49     // num buckets
#define BHN  64     // B*H
#define HN   8      // H
#define WTS  66     // LDS row stride for transposed W (bank spreading, keeps pairs 8B-aligned)

typedef __attribute__((ext_vector_type(2))) float v2f;
typedef __attribute__((ext_vector_type(8))) float v8f;

// ---------------- Phase 1: WMMA GEMM into workspace ----------------
// Block = 8 waves, all sharing one (b,h). Wave w computes a full 16x64 strip
// (4 accumulators) of lt[bh] for its M-tile. K loop: 16 steps of 16x16x4.
__global__ __launch_bounds__(256) void irpe_lt_wmma_f32(
    const float* __restrict__ x,      // (BHN, LQ, DD)
    const float* __restrict__ w,      // (HN, DD, NB)
    float* __restrict__ lt)           // (BHN, LQ, NB) workspace
{
    __shared__ float s_wt[64 * WTS];  // s_wt[n*WTS + d] = W[h,d,n], zero-padded n in [49,64)

    const int tid   = threadIdx.x;
    const int lane  = tid & 31;
    const int wave  = tid >> 5;
    const int t     = blockIdx.x * 8 + wave;   // 4096 strip-waves total
    const int mtile = t & 63;                  // 64 M-tiles per (b,h)
    const int bh    = t >> 6;                  // same for all 8 waves in block
    const int h     = bh & (HN - 1);

    // Stage W[h] transposed into LDS, branchlessly zero-padded (clamped index + mask).
    const float* wh = w + (size_t)h * DD * NB;
    for (int idx = tid; idx < 64 * 64; idx += 256) {
        const int n = idx & 63;                // consecutive tid -> consecutive n (coalesced)
        const int d = idx >> 6;
        const int nsafe = (n < NB) ? n : 0;
        float v = wh[d * NB + nsafe];
        v = (n < NB) ? v : 0.0f;
        s_wt[n * WTS + d] = v;
    }
    __syncthreads();

    const int half = lane >> 4;                // 0: lanes 0-15, 1: lanes 16-31
    const int m    = mtile * 16 + (lane & 15); // A row (M)
    const int ncl  = lane & 15;                // column within an N-tile
    const float* xrow = x + ((size_t)bh * LQ + m) * DD;

    v8f acc0 = {}, acc1 = {}, acc2 = {}, acc3 = {};
    #pragma unroll
    for (int ks = 0; ks < DD; ks += 4) {
        const int ka = ks + half * 2;          // this half-wave's K pair (even -> 8B aligned)
        const v2f a  = *(const v2f*)(xrow + ka);                       // global b64
        const v2f b0 = *(const v2f*)(&s_wt[(0 * 16 + ncl) * WTS + ka]); // ds b64
        const v2f b1 = *(const v2f*)(&s_wt[(1 * 16 + ncl) * WTS + ka]);
        const v2f b2 = *(const v2f*)(&s_wt[(2 * 16 + ncl) * WTS + ka]);
        const v2f b3 = *(const v2f*)(&s_wt[(3 * 16 + ncl) * WTS + ka]);
        acc0 = __builtin_amdgcn_wmma_f32_16x16x4_f32(false, a, false, b0, (short)0, acc0, false, false);
        acc1 = __builtin_amdgcn_wmma_f32_16x16x4_f32(false, a, false, b1, (short)0, acc1, false, false);
        acc2 = __builtin_amdgcn_wmma_f32_16x16x4_f32(false, a, false, b2, (short)0, acc2, false, false);
        acc3 = __builtin_amdgcn_wmma_f32_16x16x4_f32(false, a, false, b3, (short)0, acc3, false, false);
    }

    // C/D layout: VGPR v -> row v (lanes 0-15) or v+8 (lanes 16-31), col = lane%16.
    const int rbase = mtile * 16 + (half << 3);
    float* ltb = lt + (size_t)bh * LQ * NB;
    {   // N-tiles 0..2 are fully inside NB=49
        const v8f* accs[3] = { &acc0, &acc1, &acc2 };
        #pragma unroll
        for (int nt = 0; nt < 3; ++nt) {
            float* dst = ltb + nt * 16 + ncl;
            const v8f c = *accs[nt];
            #pragma unroll
            for (int v = 0; v < 8; ++v)
                dst[(size_t)(rbase + v) * NB] = c[v];
        }
    }
    if (48 + ncl < NB) {                       // N-tile 3: only column 48 is real
        float* dst = ltb + 48 + ncl;
        #pragma unroll
        for (int v = 0; v < 8; ++v)
            dst[(size_t)(rbase + v) * NB] = acc3[v];
    }
}

// ---------------- Phase 2: HBM-saturating gather ----------------
// grid = (LQ, BHN/16), block = 256 threads. Block (i, g): stages lt rows for
// 16 (b,h) into LDS, each thread emits one float4 per (b,h) -> b128 stores.
__global__ __launch_bounds__(256) void irpe_gather_bcast(
    const float* __restrict__ lt,     // (BHN, LQ, NB)
    const int*   __restrict__ rp,     // (LQ, LQ)
    float*       __restrict__ out)    // (BHN, LQ, LQ)
{
    __shared__ float s_lt[16 * NB];   // 3136 B

    const int i   = blockIdx.x;
    const int bh0 = blockIdx.y * 16;
    const int tid = threadIdx.x;

    // Stage lt[bh0..bh0+15, i, 0..48] into LDS (784 floats, L2-hot)
    for (int idx = tid; idx < 16 * NB; idx += 256) {
        const int bh = idx / NB;
        const int k  = idx - bh * NB;
        s_lt[idx] = lt[((size_t)(bh0 + bh) * LQ + i) * NB + k];
    }
    __syncthreads();

    // Each thread owns 4 consecutive j; bucket indices via one int4 load (L2-hot)
    const int j = tid * 4;
    const int4 bk = *(const int4*)(rp + (size_t)i * LQ + j);

    #pragma unroll 4
    for (int bh = 0; bh < 16; ++bh) {
        const float* row = s_lt + bh * NB;
        float4 v;
        v.x = row[bk.x];
        v.y = row[bk.y];
        v.z = row[bk.z];
        v.w = row[bk.w];
        *(float4*)(out + (((size_t)(bh0 + bh) * LQ + i) * LQ + j)) = v;  // b128 store
    }
}

extern "C" void kernel_launch(void* const* d_in, const int* in_sizes, int n_in,
                              void* d_out, int out_size, void* d_ws, size_t ws_size,
                              hipStream_t stream) {
    const float* x   = (const float*)d_in[0];  // (8,8,1024,64) f32
    const float* w   = (const float*)d_in[1];  // (8,64,49) f32
    const int*   rp  = (const int*)d_in[2];    // (1024,1024) i32
    float*       out = (float*)d_out;          // (8,8,1024,1024) f32
    float*       lt  = (float*)d_ws;           // needs 64*1024*49*4 ~= 12.85 MB

    // Phase 1: 4096 strip-waves -> 512 blocks of 8 waves (one bh per block)
    irpe_lt_wmma_f32<<<512, 256, 0, stream>>>(x, w, lt);
    // Phase 2: 1024 x 4 blocks, each writes 64 KB of coalesced float4
    irpe_gather_bcast<<<dim3(LQ, BHN / 16), 256, 0, stream>>>(lt, rp, out);
}